// GCN_64510408786513
// MI455X (gfx1250) — compile-verified
//
#include <hip/hip_runtime.h>

typedef float v2f __attribute__((ext_vector_type(2)));
typedef float v8f __attribute__((ext_vector_type(8)));

#define H 128
#define LNEG 0.2f
#define BNEPS 1e-5f
#define NLAYER 3

__device__ __forceinline__ void atomAddF(float* p, float v) {
  unsafeAtomicAdd(p, v);  // global_atomic_add_f32
}

__device__ __forceinline__ float lrelu(float v) {
  return v >= 0.0f ? v : LNEG * v;
}

// Stage a row-major weight block (nfloats, multiple of 4) into LDS.
__device__ __forceinline__ void stageW(float* sW, const float* __restrict__ W, int nfloats) {
  for (int i = threadIdx.x; i < (nfloats >> 2); i += blockDim.x) {
    ((float4*)sW)[i] = ((const float4*)W)[i];
  }
}

// One wave computes a 16x16 tile of X[row0:row0+16, :] @ W (W 128x128 in LDS),
// with optional per-row scale on A (ascale = scale for row (row0 + lane&15)).
// Fragment layouts per CDNA5 ISA:
//   A (16x4 f32): lane m = lane&15, k = 2*(lane>>4) + vgpr  -> contiguous float2
//   B (4x16 f32): lane n = lane&15, k = 2*(lane>>4) + vgpr
__device__ __forceinline__ v8f tile_mma(const float* __restrict__ X, int row0,
                                        const float* sW, int col0, int lane,
                                        float ascale, v8f acc) {
  const int m = lane & 15;
  const int g = lane >> 4;
  const float* xr = X + (size_t)(row0 + m) * H;
#pragma unroll
  for (int kb = 0; kb < H; kb += 4) {
    const int k0 = kb + 2 * g;
    float2 av = *(const float2*)(xr + k0);
    v2f a;
    a.x = av.x * ascale;
    a.y = av.y * ascale;
    v2f b;
    b.x = sW[k0 * H + col0 + m];
    b.y = sW[(k0 + 1) * H + col0 + m];
    acc = __builtin_amdgcn_wmma_f32_16x16x4_f32(false, a, false, b, (short)0, acc,
                                                false, false);
  }
  return acc;
}

// ---------------- GEMM kernels (8 waves/block, wave w -> cols w*16..w*16+15) --------

// Y = X @ W ; Ycopy = Y (seeds S with the self-loop contribution h[i])
__global__ void k_gemm_lin(const float* __restrict__ X, const float* __restrict__ W,
                           float* __restrict__ Y, float* __restrict__ Ycopy, int nstrip) {
  __shared__ float sW[H * H];
  stageW(sW, W, H * H);
  __syncthreads();
  const int lane = threadIdx.x & 31;
  const int col0 = (threadIdx.x >> 5) * 16;
  const int n = lane & 15, g = lane >> 4;
  for (int s = blockIdx.x; s < nstrip; s += gridDim.x) {
    const int row0 = s * 16;
    v8f acc = {};
    acc = tile_mma(X, row0, sW, col0, lane, 1.0f, acc);
#pragma unroll
    for (int v = 0; v < 8; ++v) {
      const size_t off = (size_t)(row0 + v + 8 * g) * H + col0 + n;
      Y[off] = acc[v];
      Ycopy[off] = acc[v];
    }
  }
}

// agg = (indeg .* h) @ We_top + indeg*be + bg
__global__ void k_gemm_agg1(const float* __restrict__ hmat, const float* __restrict__ Wtop,
                            const float* __restrict__ be, const float* __restrict__ bg,
                            const float* __restrict__ indeg, float* __restrict__ agg,
                            int nstrip) {
  __shared__ float sW[H * H];
  stageW(sW, Wtop, H * H);
  __syncthreads();
  const int lane = threadIdx.x & 31;
  const int col0 = (threadIdx.x >> 5) * 16;
  const int n = lane & 15, g = lane >> 4;
  const float bev = be[col0 + n];
  const float bgv = bg[col0 + n];
  for (int s = blockIdx.x; s < nstrip; s += gridDim.x) {
    const int row0 = s * 16;
    const float dm = indeg[row0 + n];  // scale for A row m = lane&15 (== n)
    v8f acc;
#pragma unroll
    for (int v = 0; v < 8; ++v) acc[v] = indeg[row0 + v + 8 * g] * bev + bgv;
    acc = tile_mma(hmat, row0, sW, col0, lane, dm, acc);
#pragma unroll
    for (int v = 0; v < 8; ++v)
      agg[(size_t)(row0 + v + 8 * g) * H + col0 + n] = acc[v];
  }
}

// agg += S @ We_bot
__global__ void k_gemm_agg2(const float* __restrict__ S, const float* __restrict__ Wbot,
                            float* __restrict__ agg, int nstrip) {
  __shared__ float sW[H * H];
  stageW(sW, Wbot, H * H);
  __syncthreads();
  const int lane = threadIdx.x & 31;
  const int col0 = (threadIdx.x >> 5) * 16;
  const int n = lane & 15, g = lane >> 4;
  for (int s = blockIdx.x; s < nstrip; s += gridDim.x) {
    const int row0 = s * 16;
    v8f acc;
#pragma unroll
    for (int v = 0; v < 8; ++v)
      acc[v] = agg[(size_t)(row0 + v + 8 * g) * H + col0 + n];
    acc = tile_mma(S, row0, sW, col0, lane, 1.0f, acc);
#pragma unroll
    for (int v = 0; v < 8; ++v)
      agg[(size_t)(row0 + v + 8 * g) * H + col0 + n] = acc[v];
  }
}

// h2 = lrelu(agg @ Wl + bl); accumulate per-channel sum / sumsq for BatchNorm.
__global__ void k_gemm_post(const float* __restrict__ agg, const float* __restrict__ Wl,
                            const float* __restrict__ bl, float* __restrict__ h2,
                            float* __restrict__ stats, int nstrip) {
  __shared__ float sW[H * H];
  __shared__ float ls[2 * H];
  stageW(sW, Wl, H * H);
  for (int i = threadIdx.x; i < 2 * H; i += blockDim.x) ls[i] = 0.0f;
  __syncthreads();
  const int lane = threadIdx.x & 31;
  const int col0 = (threadIdx.x >> 5) * 16;
  const int n = lane & 15, g = lane >> 4;
  const float blv = bl[col0 + n];
  float psum = 0.0f, psq = 0.0f;
  for (int s = blockIdx.x; s < nstrip; s += gridDim.x) {
    const int row0 = s * 16;
    v8f acc;
#pragma unroll
    for (int v = 0; v < 8; ++v) acc[v] = blv;
    acc = tile_mma(agg, row0, sW, col0, lane, 1.0f, acc);
#pragma unroll
    for (int v = 0; v < 8; ++v) {
      float y = lrelu(acc[v]);
      psum += y;
      psq += y * y;
      h2[(size_t)(row0 + v + 8 * g) * H + col0 + n] = y;
    }
  }
  atomicAdd(&ls[col0 + n], psum);       // ds_add_f32 (2-way contention max)
  atomicAdd(&ls[H + col0 + n], psq);
  __syncthreads();
  for (int i = threadIdx.x; i < 2 * H; i += blockDim.x) atomAddF(&stats[i], ls[i]);
}

// ---------------- sparse / elementwise kernels -------------------------------------

__global__ void k_indeg_init(float* __restrict__ indeg, int n) {
  int t = blockIdx.x * blockDim.x + threadIdx.x;
  if (t < n) indeg[t] = 1.0f;  // self loop
}

__global__ void k_indeg_count(const int* __restrict__ ei, float* __restrict__ indeg, int E) {
  int t = blockIdx.x * blockDim.x + threadIdx.x;
  if (t < E) atomAddF(&indeg[ei[E + t]], 1.0f);  // col = edge_index[1]
}

// S[col[e]] += h[row[e]] : 32 threads per edge, float4 per thread
__global__ void k_scatter(const int* __restrict__ ei, const float* __restrict__ hmat,
                          float* __restrict__ S, int E) {
  int t = blockIdx.x * blockDim.x + threadIdx.x;
  int e = t >> 5;
  if (e >= E) return;
  int c = (t & 31) * 4;
  int src = ei[e];
  int dst = ei[E + e];
  const float4 v = *(const float4*)(hmat + (size_t)src * H + c);
  float* d = S + (size_t)dst * H + c;
  atomAddF(d + 0, v.x);
  atomAddF(d + 1, v.y);
  atomAddF(d + 2, v.z);
  atomAddF(d + 3, v.w);
}

__global__ void k_bn(const float* __restrict__ h2, const float* __restrict__ stats,
                     const float* __restrict__ gamma, const float* __restrict__ beta,
                     float* __restrict__ z, int n_nodes) {
  int t = blockIdx.x * blockDim.x + threadIdx.x;
  if (t >= n_nodes * H) return;
  int c = t & (H - 1);
  float inv_n = 1.0f / (float)n_nodes;
  float mu = stats[c] * inv_n;
  float var = stats[H + c] * inv_n - mu * mu;
  float v = h2[t];
  z[t] = gamma[c] * (v - mu) * rsqrtf(var + BNEPS) + beta[c];
}

__global__ void k_zero(float* __restrict__ p, int n) {
  int t = blockIdx.x * blockDim.x + threadIdx.x;
  if (t < n) p[t] = 0.0f;
}

__global__ void k_pool(const float* __restrict__ z, const int* __restrict__ batch,
                       float* __restrict__ psum, float* __restrict__ cnt, int n_nodes) {
  int t = blockIdx.x * blockDim.x + threadIdx.x;
  if (t >= n_nodes * H) return;
  int i = t >> 7, c = t & (H - 1);
  int b = batch[i];
  atomAddF(&psum[b * H + c], z[t]);
  if (c == 0) atomAddF(&cnt[b], 1.0f);
}

// Final tiny MLP head: one block per graph (128 threads)
__global__ void k_head(const float* __restrict__ psum, const float* __restrict__ cnt,
                       const float* __restrict__ Wf1, const float* __restrict__ bf1,
                       const float* __restrict__ Wf2, const float* __restrict__ bf2,
                       const float* __restrict__ Wf3, const float* __restrict__ bf3,
                       float* __restrict__ out) {
  __shared__ float p[H], q[H], r[H / 2];
  const int g = blockIdx.x, c = threadIdx.x;
  float cn = cnt[g];
  cn = cn > 1.0f ? cn : 1.0f;
  p[c] = psum[g * H + c] / cn;
  __syncthreads();
  float a = bf1[c];
  for (int k = 0; k < H; ++k) a += p[k] * Wf1[k * H + c];
  q[c] = lrelu(a);
  __syncthreads();
  if (c < H / 2) {
    float b = bf2[c];
    for (int k = 0; k < H; ++k) b += q[k] * Wf2[k * (H / 2) + c];
    r[c] = lrelu(b);
  }
  __syncthreads();
  if (c < 2) {
    float o = bf3[c];
    for (int k = 0; k < H / 2; ++k) o += r[k] * Wf3[k * 2 + c];
    out[g * 2 + c] = o;
  }
}

extern "C" void kernel_launch(void* const* d_in, const int* in_sizes, int n_in,
                              void* d_out, int out_size, void* d_ws, size_t ws_size,
                              hipStream_t stream) {
  const float* x     = (const float*)d_in[0];
  const int*   ei    = (const int*)d_in[1];
  const int*   batch = (const int*)d_in[3];
  const float* Wg    = (const float*)d_in[4];
  const float* bg    = (const float*)d_in[5];
  const float* We    = (const float*)d_in[6];
  const float* be    = (const float*)d_in[7];
  const float* Wl    = (const float*)d_in[8];
  const float* bl    = (const float*)d_in[9];
  const float* gamma = (const float*)d_in[10];
  const float* beta  = (const float*)d_in[11];
  const float* Wf1 = (const float*)d_in[12];
  const float* bf1 = (const float*)d_in[13];
  const float* Wf2 = (const float*)d_in[14];
  const float* bf2 = (const float*)d_in[15];
  const float* Wf3 = (const float*)d_in[16];
  const float* bf3 = (const float*)d_in[17];
  float* out = (float*)d_out;
  float* ws = (float*)d_ws;

  const int N = in_sizes[0] / H;   // 50000
  const int E = in_sizes[1] / 2;   // 800000
  const int B = 50;
  const int nstrip = N / 16;       // 3125 (N is a multiple of 16)

  // workspace layout (floats, 64-float aligned chunks)
  size_t o = 0;
  float* indeg = ws + o; o += ((size_t)N + 63) & ~(size_t)63;
  float* hbuf  = ws + o; o += (size_t)N * H;   // also reused for h2
  float* Sbuf  = ws + o; o += (size_t)N * H;
  float* aggb  = ws + o; o += (size_t)N * H;
  float* zbuf  = ws + o; o += (size_t)N * H;
  float* stats = ws + o; o += 256;             // [sum(128), sumsq(128)]
  float* psum  = ws + o; o += (size_t)B * H;
  float* cnt   = ws + o; o += 64;              // contiguous with psum

  const dim3 blk(256);
  const int GB = 512;  // persistent-ish GEMM grid

  k_indeg_init<<<(N + 255) / 256, blk, 0, stream>>>(indeg, N);
  k_indeg_count<<<(E + 255) / 256, blk, 0, stream>>>(ei, indeg, E);
  k_zero<<<(B * H + 64 + 255) / 256, blk, 0, stream>>>(psum, B * H + 64);

  for (int l = 0; l < NLAYER; ++l) {
    const float* Xin = (l == 0) ? x : zbuf;
    const float* Wel = We + (size_t)l * 2 * H * H;
    k_zero<<<1, 256, 0, stream>>>(stats, 256);
    // h = z @ Wg[l]; S = h (self-loop seed)
    k_gemm_lin<<<GB, blk, 0, stream>>>(Xin, Wg + (size_t)l * H * H, hbuf, Sbuf, nstrip);
    // S[col] += h[row] over all edges
    {
      long long tot = (long long)E * 32;
      k_scatter<<<(int)((tot + 255) / 256), blk, 0, stream>>>(ei, hbuf, Sbuf, E);
    }
    // agg = (indeg .* h) @ We_top + indeg*be + bg
    k_gemm_agg1<<<GB, blk, 0, stream>>>(hbuf, Wel, be + l * H, bg + l * H, indeg,
                                        aggb, nstrip);
    // agg += S @ We_bot
    k_gemm_agg2<<<GB, blk, 0, stream>>>(Sbuf, Wel + H * H, aggb, nstrip);
    // h2 = lrelu(agg @ Wl + bl) + BN stats  (h2 reuses hbuf)
    k_gemm_post<<<GB, blk, 0, stream>>>(aggb, Wl + (size_t)l * H * H, bl + l * H,
                                        hbuf, stats, nstrip);
    // z = BN(h2)
    k_bn<<<(N * H + 255) / 256, blk, 0, stream>>>(hbuf, stats, gamma + l * H,
                                                  beta + l * H, zbuf, N);
  }

  k_pool<<<(N * H + 255) / 256, blk, 0, stream>>>(zbuf, batch, psum, cnt, N);
  k_head<<<B, H, 0, stream>>>(psum, cnt, Wf1, bf1, Wf2, bf2, Wf3, bf3, out);
}